// Net_28424093565757
// MI455X (gfx1250) — compile-verified
//
#include <hip/hip_runtime.h>

#define N_NODES 100000
#define EDGES   1600000
#define DIM     64
#define TILES_PER_BLOCK 4
#define BLOCK_THREADS   128

typedef float v2f __attribute__((ext_vector_type(2)));
typedef float v8f __attribute__((ext_vector_type(8)));
typedef int   v4i __attribute__((ext_vector_type(4)));

// ---------------------------------------------------------------------------
// Async global->LDS staging (CDNA5 GLOBAL_LOAD_ASYNC_TO_LDS_B128, ASYNCcnt).
// Builtin signature (from hipcc diagnostic): param0 = int4 addrspace(1)*
// (HIP prints addrspace(1) as __device__), param1 = LDS side, then imm
// offset + cpol. Falls back to synchronous LDS stores if unavailable.
// ---------------------------------------------------------------------------
__device__ __forceinline__ void stage_f4(const float4* __restrict__ g,
                                         float4* l) {
#if __has_builtin(__builtin_amdgcn_global_load_async_to_lds_b128)
    __builtin_amdgcn_global_load_async_to_lds_b128(
        (__attribute__((address_space(1))) v4i*)g,
        (__attribute__((address_space(3))) v4i*)l,
        0, 0);
#else
    *l = *g;
#endif
}

__device__ __forceinline__ void stage_wait() {
#if __has_builtin(__builtin_amdgcn_global_load_async_to_lds_b128)
#if __has_builtin(__builtin_amdgcn_s_wait_asynccnt)
    __builtin_amdgcn_s_wait_asynccnt(0);
#else
    asm volatile("s_wait_asynccnt 0x0" ::: "memory");
#endif
#endif
}

// ---------------------------------------------------------------------------
// Scatter-add: agg[dst] += x[src] over all edges. 16 threads per edge,
// float4 gather + 4 scalar f32 atomics (global_atomic_add_f32, L2-resident).
// ---------------------------------------------------------------------------
__global__ __launch_bounds__(256) void scatter_add_kernel(
    const float* __restrict__ x,
    const long long* __restrict__ ei,   // int64 [2, E]
    float* __restrict__ agg) {
    int tid = blockIdx.x * blockDim.x + threadIdx.x;
    int e = tid >> 4;
    int c = tid & 15;
    if (e >= EDGES) return;
    long long s = ei[e];
    long long d = ei[EDGES + e];
    const float4 v = ((const float4*)(x + (size_t)s * DIM))[c];
    float* out = agg + (size_t)d * DIM + c * 4;
    atomicAdd(out + 0, v.x);
    atomicAdd(out + 1, v.y);
    atomicAdd(out + 2, v.z);
    atomicAdd(out + 3, v.w);
}

// ---------------------------------------------------------------------------
// WMMA fragment builders for V_WMMA_F32_16X16X4_F32.
// A (16x4 f32): lanes 0-15 -> M=lane, VGPR0=K0, VGPR1=K1;
//               lanes 16-31 -> M=lane-16, VGPR0=K2, VGPR1=K3.
// B (4x16 f32): mirrored (lanes 0-15: K=kb,kb+1; lanes 16-31: K=kb+2,kb+3),
//               N = column striped across lanes.
// ---------------------------------------------------------------------------
__device__ __forceinline__ v2f frag_a_lds(const float* hsrow_base, int lane,
                                          int kc, int row_stride) {
    int row = lane & 15;
    int kb  = 4 * kc + ((lane >> 4) << 1);
    const float* p = hsrow_base + row * row_stride + kb;
    v2f a;
    a.x = p[0];
    a.y = p[1];
    return a;
}

__device__ __forceinline__ v2f frag_b_lds(const float* W, int lane,
                                          int kc, int nt) {
    int n  = nt * 16 + (lane & 15);
    int kb = 4 * kc + ((lane >> 4) << 1);
    v2f b;
    b.x = W[kb * DIM + n];
    b.y = W[(kb + 1) * DIM + n];
    return b;
}

// ---------------------------------------------------------------------------
// Fused GIN MLP: out = relu( relu(h @ w1 + b1) @ w2 + b2 )
// 128 threads = 4 waves; weights staged once per block into LDS (async),
// each wave owns one 16x64 node tile staged via coalesced b128 loads.
// Intermediate h1 round-trips through the wave-private LDS tile (no barrier
// needed: DS ops are in-order within a wave).
// ---------------------------------------------------------------------------
__global__ __launch_bounds__(BLOCK_THREADS) void gin_mlp_kernel(
    const float* __restrict__ h,    // agg = sum_j x_j + x_i, [N, 64]
    const float* __restrict__ w1, const float* __restrict__ b1,
    const float* __restrict__ w2, const float* __restrict__ b2,
    float* __restrict__ out) {
    __shared__ float ws1[DIM * DIM];                       // 16 KB
    __shared__ float ws2[DIM * DIM];                       // 16 KB
    __shared__ float hs[TILES_PER_BLOCK][16][DIM + 4];     // 17 KB (padded)

    const int t    = threadIdx.x;
    const int lane = t & 31;
    const int w    = t >> 5;
    const int tile = blockIdx.x * TILES_PER_BLOCK + w;
    const bool active = (tile * 16 < N_NODES);             // wave-uniform

    // Stage both weight matrices: 2 x 1024 float4 over 128 threads (async).
    {
        const float4* g1 = (const float4*)w1;
        const float4* g2 = (const float4*)w2;
        float4* l1 = (float4*)ws1;
        float4* l2 = (float4*)ws2;
        #pragma unroll
        for (int i = 0; i < 8; ++i) {
            stage_f4(g1 + t + BLOCK_THREADS * i, l1 + t + BLOCK_THREADS * i);
            stage_f4(g2 + t + BLOCK_THREADS * i, l2 + t + BLOCK_THREADS * i);
        }
    }

    // Stage this wave's 16x64 A-tile with coalesced b128 loads.
    if (active) {
        const float4* A4 = (const float4*)(h + (size_t)tile * 16 * DIM);
        #pragma unroll
        for (int i = 0; i < 8; ++i) {
            int idx = lane + 32 * i;        // 0..255 float4s
            int row = idx >> 4;
            int c4  = idx & 15;
            *(float4*)&hs[w][row][c4 * 4] = A4[idx];
        }
    }
    stage_wait();
    __syncthreads();
    if (!active) return;

    const int row_stride = DIM + 4;
    const float* myhs = &hs[w][0][0];
    const int mbase = (lane >> 4) << 3;     // 0 or 8

    // GEMM1: (16x64) @ w1, K in 16 chunks of 4
    v8f acc[4];
    #pragma unroll
    for (int nt = 0; nt < 4; ++nt) {
        v8f c = {};
        #pragma unroll
        for (int kc = 0; kc < 16; ++kc) {
            v2f a = frag_a_lds(myhs, lane, kc, row_stride);
            v2f b = frag_b_lds(ws1, lane, kc, nt);
            c = __builtin_amdgcn_wmma_f32_16x16x4_f32(
                    false, a, false, b, (short)0, c, false, false);
        }
        acc[nt] = c;
    }

    // bias + relu, deposit h1 back into this wave's LDS tile
    #pragma unroll
    for (int nt = 0; nt < 4; ++nt) {
        int n = nt * 16 + (lane & 15);
        float bias = b1[n];
        #pragma unroll
        for (int r = 0; r < 8; ++r) {
            float v = acc[nt][r] + bias;
            hs[w][mbase + r][n] = v > 0.0f ? v : 0.0f;
        }
    }

    // GEMM2: h1 @ w2, bias + relu, store
    float* O = out + (size_t)tile * 16 * DIM;
    #pragma unroll
    for (int nt = 0; nt < 4; ++nt) {
        v8f c = {};
        #pragma unroll
        for (int kc = 0; kc < 16; ++kc) {
            v2f a = frag_a_lds(myhs, lane, kc, row_stride);
            v2f b = frag_b_lds(ws2, lane, kc, nt);
            c = __builtin_amdgcn_wmma_f32_16x16x4_f32(
                    false, a, false, b, (short)0, c, false, false);
        }
        int n = nt * 16 + (lane & 15);
        float bias = b2[n];
        #pragma unroll
        for (int r = 0; r < 8; ++r) {
            float v = c[r] + bias;
            O[(mbase + r) * DIM + n] = v > 0.0f ? v : 0.0f;
        }
    }
}

// ---------------------------------------------------------------------------
// Final projection: out = x @ wf + bf   (no relu)
// ---------------------------------------------------------------------------
__global__ __launch_bounds__(BLOCK_THREADS) void final_gemm_kernel(
    const float* __restrict__ x,
    const float* __restrict__ wf, const float* __restrict__ bf,
    float* __restrict__ out) {
    __shared__ float wsf[DIM * DIM];
    __shared__ float hs[TILES_PER_BLOCK][16][DIM + 4];

    const int t    = threadIdx.x;
    const int lane = t & 31;
    const int w    = t >> 5;
    const int tile = blockIdx.x * TILES_PER_BLOCK + w;
    const bool active = (tile * 16 < N_NODES);

    {
        const float4* g = (const float4*)wf;
        float4* l = (float4*)wsf;
        #pragma unroll
        for (int i = 0; i < 8; ++i)
            stage_f4(g + t + BLOCK_THREADS * i, l + t + BLOCK_THREADS * i);
    }
    if (active) {
        const float4* A4 = (const float4*)(x + (size_t)tile * 16 * DIM);
        #pragma unroll
        for (int i = 0; i < 8; ++i) {
            int idx = lane + 32 * i;
            *(float4*)&hs[w][idx >> 4][(idx & 15) * 4] = A4[idx];
        }
    }
    stage_wait();
    __syncthreads();
    if (!active) return;

    const int row_stride = DIM + 4;
    const float* myhs = &hs[w][0][0];
    const int mbase = (lane >> 4) << 3;
    float* O = out + (size_t)tile * 16 * DIM;

    #pragma unroll
    for (int nt = 0; nt < 4; ++nt) {
        v8f c = {};
        #pragma unroll
        for (int kc = 0; kc < 16; ++kc) {
            v2f a = frag_a_lds(myhs, lane, kc, row_stride);
            v2f b = frag_b_lds(wsf, lane, kc, nt);
            c = __builtin_amdgcn_wmma_f32_16x16x4_f32(
                    false, a, false, b, (short)0, c, false, false);
        }
        int n = nt * 16 + (lane & 15);
        float bias = bf[n];
        #pragma unroll
        for (int r = 0; r < 8; ++r) {
            O[(mbase + r) * DIM + n] = c[r] + bias;
        }
    }
}

// ---------------------------------------------------------------------------
// Driver
// inputs: [0]=x, [1]=edge_index(int64),
//         [2..5]=w1_0,b1_0,w2_0,b2_0, [6..9]=block1, [10..13]=block2,
//         [14]=wf, [15]=bf
// workspace: agg (N*D) | buf0 (N*D) | buf1 (N*D)   = 76.8 MB
// ---------------------------------------------------------------------------
extern "C" void kernel_launch(void* const* d_in, const int* in_sizes, int n_in,
                              void* d_out, int out_size, void* d_ws, size_t ws_size,
                              hipStream_t stream) {
    const float*     x  = (const float*)d_in[0];
    const long long* ei = (const long long*)d_in[1];

    float* agg  = (float*)d_ws;
    float* buf0 = agg  + (size_t)N_NODES * DIM;
    float* buf1 = buf0 + (size_t)N_NODES * DIM;

    const size_t feat_bytes = (size_t)N_NODES * DIM * sizeof(float);
    const int scatter_blocks = (EDGES * 16) / 256;                       // exact
    const int mlp_blocks =
        (N_NODES / 16 + TILES_PER_BLOCK - 1) / TILES_PER_BLOCK;          // 1563

    const float* cur = x;
    float* bufs[3] = {buf0, buf1, buf0};
    for (int b = 0; b < 3; ++b) {
        const float* w1 = (const float*)d_in[2 + 4 * b + 0];
        const float* b1 = (const float*)d_in[2 + 4 * b + 1];
        const float* w2 = (const float*)d_in[2 + 4 * b + 2];
        const float* b2 = (const float*)d_in[2 + 4 * b + 3];
        float* nxt = bufs[b];

        // agg = x  (the "+ (1+eps)*x_i" self term, eps=0)
        (void)hipMemcpyAsync(agg, cur, feat_bytes, hipMemcpyDeviceToDevice, stream);
        // agg[dst] += x[src] over all edges
        scatter_add_kernel<<<scatter_blocks, 256, 0, stream>>>(cur, ei, agg);
        // MLP + relus
        gin_mlp_kernel<<<mlp_blocks, BLOCK_THREADS, 0, stream>>>(
            agg, w1, b1, w2, b2, nxt);
        cur = nxt;
    }

    final_gemm_kernel<<<mlp_blocks, BLOCK_THREADS, 0, stream>>>(
        cur, (const float*)d_in[14], (const float*)d_in[15], (float*)d_out);
}